// LowrankLearnableHash_584115553031
// MI455X (gfx1250) — compile-verified
//
#include <hip/hip_runtime.h>
#include <stdint.h>

#define NPTS      (1u << 21)         // 2,097,152 points
#define PRES      512
#define CPLANE    18
#define CPAD      20                 // pad 18 -> 20 floats: taps are 5 aligned float4
#define FRES      128
#define FC        32                 // feature channels == wave32 width
#define FVOX      (FRES * FRES * FRES)

#define PLANE_T_FLOATS  (PRES * PRES * CPAD)      // 5,242,880
#define PLANES_T_FLOATS (3u * PLANE_T_FLOATS)
#define FEAT_T_FLOATS   ((size_t)FVOX * FC)       // 67,108,864

__device__ __forceinline__ int imin(int a, int b) { return a < b ? a : b; }

// ---------------------------------------------------------------------------
// Transpose planes [C=18,512,512] -> [512*512, CPAD=20] (channel-last, padded)
// Read-once input -> non-temporal loads (TH=NT) to keep L2 for hot data.
// ---------------------------------------------------------------------------
__global__ __launch_bounds__(256) void k_transpose_planes(
    const float* __restrict__ p0, const float* __restrict__ p1,
    const float* __restrict__ p2, float* __restrict__ outT) {
  const int plane = blockIdx.y;
  const float* in = (plane == 0) ? p0 : ((plane == 1) ? p1 : p2);
  unsigned t = blockIdx.x * 256u + threadIdx.x;          // t = v*CPAD + c
  unsigned c = t % CPAD;
  unsigned v = t / CPAD;
  float val = (c < CPLANE) ? __builtin_nontemporal_load(&in[(size_t)c * (PRES * PRES) + v])
                           : 0.0f;
  outT[(size_t)plane * PLANE_T_FLOATS + t] = val;        // fully coalesced store
}

// ---------------------------------------------------------------------------
// Transpose features [C=32, 128^3] -> [128^3, 32] via LDS 32x32 tile.
// Source is read-once -> NT loads; dest (featT) is the hot L2 working set ->
// regular-temporal stores.
// ---------------------------------------------------------------------------
__global__ __launch_bounds__(256) void k_transpose_feat(
    const float* __restrict__ in, float* __restrict__ outT) {
  __shared__ float s[32][33];                            // +1 pad: no bank conflicts
  const unsigned v0 = blockIdx.x * 32u;
  const unsigned l = threadIdx.x & 31u;
  const unsigned w = threadIdx.x >> 5;
#pragma unroll
  for (int i = 0; i < 4; i++) {
    unsigned c = w + i * 8;
    s[c][l] = __builtin_nontemporal_load(&in[(size_t)c * FVOX + v0 + l]);
  }
  __syncthreads();
#pragma unroll
  for (int i = 0; i < 4; i++) {
    unsigned e = i * 256u + threadIdx.x;                 // e = vi*32 + c
    unsigned vi = e >> 5, c = e & 31u;
    outT[(size_t)v0 * 32 + e] = s[c][vi];                // coalesced stores
  }
}

// ---------------------------------------------------------------------------
// Phase 1: one point per lane. Three bilinear plane taps (channel-last, 5x
// float4 per tap, planes are L2-resident), rank-6 reduction -> interp coords.
// ---------------------------------------------------------------------------
__global__ __launch_bounds__(256) void k_interp(
    const float* __restrict__ pts, const float* __restrict__ planesT,
    float* __restrict__ interp) {
  const unsigned n = blockIdx.x * 256u + threadIdx.x;
  const float px = __builtin_nontemporal_load(&pts[(size_t)n * 3 + 0]);
  const float py = __builtin_nontemporal_load(&pts[(size_t)n * 3 + 1]);
  const float pz = __builtin_nontemporal_load(&pts[(size_t)n * 3 + 2]);

  float prod[CPAD];
  const float uu[3] = {px, px, py};
  const float vv[3] = {py, pz, pz};

#pragma unroll
  for (int p = 0; p < 3; p++) {
    float gx = fminf(fmaxf((uu[p] + 1.0f) * 0.5f * 511.0f, 0.0f), 511.0f);
    float gy = fminf(fmaxf((vv[p] + 1.0f) * 0.5f * 511.0f, 0.0f), 511.0f);
    int ix0 = imin((int)gx, 511);
    int iy0 = imin((int)gy, 511);
    int ix1 = imin(ix0 + 1, 511);
    int iy1 = imin(iy0 + 1, 511);
    float fx = gx - (float)ix0;
    float fy = gy - (float)iy0;
    float w00 = (1.0f - fx) * (1.0f - fy), w01 = fx * (1.0f - fy);
    float w10 = (1.0f - fx) * fy,          w11 = fx * fy;

    const float* base = planesT + (size_t)p * PLANE_T_FLOATS;
    const float4* A = (const float4*)(base + (size_t)(iy0 * PRES + ix0) * CPAD);
    const float4* B = (const float4*)(base + (size_t)(iy0 * PRES + ix1) * CPAD);
    const float4* C = (const float4*)(base + (size_t)(iy1 * PRES + ix0) * CPAD);
    const float4* D = (const float4*)(base + (size_t)(iy1 * PRES + ix1) * CPAD);
#pragma unroll
    for (int q = 0; q < 5; q++) {
      float4 a = A[q], b = B[q], c = C[q], d = D[q];
      float r0 = w00 * a.x + w01 * b.x + w10 * c.x + w11 * d.x;
      float r1 = w00 * a.y + w01 * b.y + w10 * c.y + w11 * d.y;
      float r2 = w00 * a.z + w01 * b.z + w10 * c.z + w11 * d.z;
      float r3 = w00 * a.w + w01 * b.w + w10 * c.w + w11 * d.w;
      if (p == 0) {
        prod[q * 4 + 0] = r0; prod[q * 4 + 1] = r1;
        prod[q * 4 + 2] = r2; prod[q * 4 + 3] = r3;
      } else {
        prod[q * 4 + 0] *= r0; prod[q * 4 + 1] *= r1;
        prod[q * 4 + 2] *= r2; prod[q * 4 + 3] *= r3;
      }
    }
  }
  float o0 = 0.f, o1 = 0.f, o2 = 0.f;
#pragma unroll
  for (int r = 0; r < 6; r++) {
    o0 += prod[r]; o1 += prod[6 + r]; o2 += prod[12 + r];
  }
  interp[(size_t)n * 3 + 0] = o0;
  interp[(size_t)n * 3 + 1] = o1;
  interp[(size_t)n * 3 + 2] = o2;
}

// ---------------------------------------------------------------------------
// Phase 2: wave-cooperative trilinear sampling, lane == channel (wave32 == 32
// channels). Coords double-buffered via async global->LDS (next chunk's
// staging overlaps the current FMA sweep); per-point params in LDS; feature
// cachelines warmed with global_prefetch_b8; output written non-temporal so
// the 268MB result stream does not evict the L2-resident feature volume.
// ---------------------------------------------------------------------------
#define CHUNK 32
#define WPB   8                                   // waves per 256-thread block
#define SWF   (96 * 2 + 12 * CHUNK)               // 2 coord bufs + params = 576 floats

__global__ __launch_bounds__(256) void k_feat_sample(
    const float* __restrict__ interp, const float* __restrict__ featT,
    float* __restrict__ out, unsigned nChunks) {
  __shared__ float smem[WPB * SWF];
  const unsigned lane = threadIdx.x & 31u;
  const unsigned wib = threadIdx.x >> 5;
  const unsigned wglob = blockIdx.x * WPB + wib;
  const unsigned nwaves = gridDim.x * WPB;

  float* sw = smem + wib * SWF;
  float* scoord = sw;                 // 2 x 96 floats: coords double buffer
  float* sparam = sw + 192;           // 32 * 12: {v,dx,dy,dz, w0..w7}
  int* iparam = (int*)sparam;
  const uint32_t ldsbase = wib * (SWF * 4u);   // smem is sole LDS object -> offset 0

  unsigned chunk = wglob;
  int buf = 0;

  // prologue: stage first chunk's coords
  if (chunk < nChunks) {
    const float* gsrc = interp + (size_t)chunk * CHUNK * 3;
#pragma unroll
    for (int r = 0; r < 3; r++) {
      uint64_t ga = (uint64_t)(uintptr_t)(gsrc + r * 32 + lane);
      uint32_t la = ldsbase + (r * 32u + lane) * 4u;
      asm volatile("global_load_async_to_lds_b32 %0, %1, off"
                   :: "v"(la), "v"(ga) : "memory");
    }
  }

  for (; chunk < nChunks; chunk += nwaves) {
    const size_t pbase = (size_t)chunk * CHUNK;

    asm volatile("s_wait_asynccnt 0x0" ::: "memory");
    const float* sc = scoord + buf * 96;

    // --- lane l computes trilinear params for point (pbase + l) ---
    float cx = sc[lane * 3 + 0];
    float cy = sc[lane * 3 + 1];
    float cz = sc[lane * 3 + 2];
    float gx = fminf(fmaxf((cx + 1.0f) * 0.5f * 127.0f, 0.0f), 127.0f);
    float gy = fminf(fmaxf((cy + 1.0f) * 0.5f * 127.0f, 0.0f), 127.0f);
    float gz = fminf(fmaxf((cz + 1.0f) * 0.5f * 127.0f, 0.0f), 127.0f);
    int ix0 = imin((int)gx, 127), iy0 = imin((int)gy, 127), iz0 = imin((int)gz, 127);
    float fx = gx - (float)ix0, fy = gy - (float)iy0, fz = gz - (float)iz0;
    int odx = (ix0 < 127) ? 1 : 0;
    int ody = ((iy0 < 127) ? 1 : 0) * FRES;
    int odz = ((iz0 < 127) ? 1 : 0) * (FRES * FRES);
    int v = (iz0 * FRES + iy0) * FRES + ix0;
    float wx0 = 1.0f - fx, wy0 = 1.0f - fy, wz0 = 1.0f - fz;

    iparam[lane * 12 + 0] = v;
    iparam[lane * 12 + 1] = odx;
    iparam[lane * 12 + 2] = ody;
    iparam[lane * 12 + 3] = odz;
    sparam[lane * 12 + 4]  = wx0 * wy0 * wz0;
    sparam[lane * 12 + 5]  = fx  * wy0 * wz0;
    sparam[lane * 12 + 6]  = wx0 * fy  * wz0;
    sparam[lane * 12 + 7]  = fx  * fy  * wz0;
    sparam[lane * 12 + 8]  = wx0 * wy0 * fz;
    sparam[lane * 12 + 9]  = fx  * wy0 * fz;
    sparam[lane * 12 + 10] = wx0 * fy  * fz;
    sparam[lane * 12 + 11] = fx  * fy  * fz;

    // --- issue next chunk's async coord staging (overlaps the sweep below) ---
    unsigned nxt = chunk + nwaves;
    if (nxt < nChunks) {
      const float* gsrc = interp + (size_t)nxt * CHUNK * 3;
      uint32_t lbuf = ldsbase + (unsigned)(buf ^ 1) * 384u;
#pragma unroll
      for (int r = 0; r < 3; r++) {
        uint64_t ga = (uint64_t)(uintptr_t)(gsrc + r * 32 + lane);
        uint32_t la = lbuf + (r * 32u + lane) * 4u;
        asm volatile("global_load_async_to_lds_b32 %0, %1, off"
                     :: "v"(la), "v"(ga) : "memory");
      }
    }

    // --- warm the 8 feature cachelines of this lane's point (gfx1250) ---
#pragma unroll
    for (int k = 0; k < 8; k++) {
      int sel = ((k & 1) ? odx : 0) + ((k & 2) ? ody : 0) + ((k & 4) ? odz : 0);
      __builtin_prefetch(featT + (size_t)(v + sel) * FC, 0, 1);
    }

    // --- sweep the 32 points; lane == channel => every tap is one 128B line ---
#pragma unroll 4
    for (int j = 0; j < CHUNK; j++) {
      int vj = iparam[j * 12 + 0];
      int dxj = iparam[j * 12 + 1];
      int dyj = iparam[j * 12 + 2];
      int dzj = iparam[j * 12 + 3];
      const float* fb = featT + (size_t)vj * FC + lane;
      float t0 = fb[0];
      float t1 = fb[(size_t)dxj * FC];
      float t2 = fb[(size_t)dyj * FC];
      float t3 = fb[(size_t)(dxj + dyj) * FC];
      float t4 = fb[(size_t)dzj * FC];
      float t5 = fb[(size_t)(dxj + dzj) * FC];
      float t6 = fb[(size_t)(dyj + dzj) * FC];
      float t7 = fb[(size_t)(dxj + dyj + dzj) * FC];
      float acc = sparam[j * 12 + 4] * t0;
      acc = fmaf(sparam[j * 12 + 5], t1, acc);
      acc = fmaf(sparam[j * 12 + 6], t2, acc);
      acc = fmaf(sparam[j * 12 + 7], t3, acc);
      acc = fmaf(sparam[j * 12 + 8], t4, acc);
      acc = fmaf(sparam[j * 12 + 9], t5, acc);
      acc = fmaf(sparam[j * 12 + 10], t6, acc);
      acc = fmaf(sparam[j * 12 + 11], t7, acc);
      __builtin_nontemporal_store(acc, &out[(pbase + (size_t)j) * FC + lane]);
    }
    buf ^= 1;
  }
}

// ---------------------------------------------------------------------------
// Fallback: monolithic direct kernel on original layouts (if ws too small)
// ---------------------------------------------------------------------------
__global__ __launch_bounds__(256) void k_fallback(
    const float* __restrict__ pts, const float* __restrict__ p01,
    const float* __restrict__ p02, const float* __restrict__ p12,
    const float* __restrict__ feat, float* __restrict__ out) {
  const unsigned n = blockIdx.x * 256u + threadIdx.x;
  float px = pts[(size_t)n * 3 + 0];
  float py = pts[(size_t)n * 3 + 1];
  float pz = pts[(size_t)n * 3 + 2];
  const float* pls[3] = {p01, p02, p12};
  const float uu[3] = {px, px, py};
  const float vv[3] = {py, pz, pz};
  float prod[CPLANE];
#pragma unroll
  for (int p = 0; p < 3; p++) {
    float gx = fminf(fmaxf((uu[p] + 1.0f) * 0.5f * 511.0f, 0.0f), 511.0f);
    float gy = fminf(fmaxf((vv[p] + 1.0f) * 0.5f * 511.0f, 0.0f), 511.0f);
    int ix0 = imin((int)gx, 511), iy0 = imin((int)gy, 511);
    int ix1 = imin(ix0 + 1, 511), iy1 = imin(iy0 + 1, 511);
    float fx = gx - (float)ix0, fy = gy - (float)iy0;
    float w00 = (1 - fx) * (1 - fy), w01 = fx * (1 - fy);
    float w10 = (1 - fx) * fy, w11 = fx * fy;
    int b00 = iy0 * PRES + ix0, b01 = iy0 * PRES + ix1;
    int b10 = iy1 * PRES + ix0, b11 = iy1 * PRES + ix1;
    const float* pl = pls[p];
#pragma unroll
    for (int c = 0; c < CPLANE; c++) {
      const float* pc = pl + (size_t)c * (PRES * PRES);
      float val = w00 * pc[b00] + w01 * pc[b01] + w10 * pc[b10] + w11 * pc[b11];
      prod[c] = (p == 0) ? val : prod[c] * val;
    }
  }
  float o[3] = {0.f, 0.f, 0.f};
#pragma unroll
  for (int oo = 0; oo < 3; oo++)
#pragma unroll
    for (int r = 0; r < 6; r++) o[oo] += prod[oo * 6 + r];

  float gx = fminf(fmaxf((o[0] + 1.0f) * 0.5f * 127.0f, 0.0f), 127.0f);
  float gy = fminf(fmaxf((o[1] + 1.0f) * 0.5f * 127.0f, 0.0f), 127.0f);
  float gz = fminf(fmaxf((o[2] + 1.0f) * 0.5f * 127.0f, 0.0f), 127.0f);
  int ix0 = imin((int)gx, 127), iy0 = imin((int)gy, 127), iz0 = imin((int)gz, 127);
  int ix1 = imin(ix0 + 1, 127), iy1 = imin(iy0 + 1, 127), iz1 = imin(iz0 + 1, 127);
  float fx = gx - ix0, fy = gy - iy0, fz = gz - iz0;
  int idx[8];
  float w[8];
  for (int k = 0; k < 8; k++) {
    int xx = (k & 1) ? ix1 : ix0;
    int yy = (k & 2) ? iy1 : iy0;
    int zz = (k & 4) ? iz1 : iz0;
    idx[k] = (zz * FRES + yy) * FRES + xx;
    w[k] = ((k & 1) ? fx : 1 - fx) * ((k & 2) ? fy : 1 - fy) * ((k & 4) ? fz : 1 - fz);
  }
#pragma unroll 4
  for (int c = 0; c < FC; c++) {
    const float* fc = feat + (size_t)c * FVOX;
    float acc = 0.f;
#pragma unroll
    for (int k = 0; k < 8; k++) acc = fmaf(w[k], fc[idx[k]], acc);
    __builtin_nontemporal_store(acc, &out[(size_t)n * FC + c]);
  }
}

// ---------------------------------------------------------------------------
extern "C" void kernel_launch(void* const* d_in, const int* in_sizes, int n_in,
                              void* d_out, int out_size, void* d_ws, size_t ws_size,
                              hipStream_t stream) {
  const float* pts = (const float*)d_in[0];
  const float* p01 = (const float*)d_in[1];
  const float* p02 = (const float*)d_in[2];
  const float* p12 = (const float*)d_in[3];
  const float* feat = (const float*)d_in[4];
  float* out = (float*)d_out;

  const size_t planesT_bytes = (size_t)PLANES_T_FLOATS * 4;     // 62,914,560
  const size_t featT_bytes = FEAT_T_FLOATS * 4;                 // 268,435,456
  const size_t interp_bytes = (size_t)NPTS * 3 * 4;             // 25,165,824
  const size_t need = planesT_bytes + featT_bytes + interp_bytes;

  if (ws_size >= need) {
    float* planesT = (float*)d_ws;
    float* featT = planesT + PLANES_T_FLOATS;
    float* interp = featT + FEAT_T_FLOATS;

    k_transpose_planes<<<dim3(PLANE_T_FLOATS / 256, 3), 256, 0, stream>>>(
        p01, p02, p12, planesT);
    k_transpose_feat<<<FVOX / 32, 256, 0, stream>>>(feat, featT);
    k_interp<<<NPTS / 256, 256, 0, stream>>>(pts, planesT, interp);
    k_feat_sample<<<2048, 256, 0, stream>>>(interp, featT, out, NPTS / CHUNK);
  } else {
    k_fallback<<<NPTS / 256, 256, 0, stream>>>(pts, p01, p02, p12, feat, out);
  }
}